// StructuredLinear_15771119911211
// MI455X (gfx1250) — compile-verified
//
#include <hip/hip_runtime.h>

typedef __attribute__((ext_vector_type(2))) float v2f;
typedef __attribute__((ext_vector_type(8))) float v8f;

#define ROWS_PER_WG 128
#define LDS_STRIDE  68   // 64 + 4 pad: 16B-aligned rows, conflict-free frag reads
#define IN_F  4096
#define N_ROWS 8192

__global__ __launch_bounds__(256)
void block_diag_wmma(const float* __restrict__ x,
                     const float* __restrict__ weight,
                     float* __restrict__ out)
{
    __shared__ float xs[ROWS_PER_WG * LDS_STRIDE];   // 128 x 64 x-strip
    __shared__ float wb[64 * LDS_STRIDE];            // 64 x 64 weight block

    const int tid  = threadIdx.x;
    const int lane = tid & 31;
    const int wave = tid >> 5;

    const int row_base = blockIdx.x * ROWS_PER_WG;
    const int b        = blockIdx.y;       // diagonal block index 0..63
    const int col_base = b * 64;

    // ---- stage x strip: 128 rows x 64 cols, coalesced b128 loads ----
    #pragma unroll
    for (int i = 0; i < 8; ++i) {
        int idx = i * 256 + tid;           // float4 index 0..2047
        int row = idx >> 4;                // 16 float4 per row
        int c4  = (idx & 15) << 2;         // float offset within row
        const float4* src = reinterpret_cast<const float4*>(
            x + (size_t)(row_base + row) * IN_F + col_base + c4);
        *reinterpret_cast<float4*>(&xs[row * LDS_STRIDE + c4]) = *src;
    }
    // ---- stage weight block: 64 x 64 (row-major, W_b[n][k]) ----
    #pragma unroll
    for (int i = 0; i < 4; ++i) {
        int idx = i * 256 + tid;           // 0..1023
        int n   = idx >> 4;
        int c4  = (idx & 15) << 2;
        const float4* src = reinterpret_cast<const float4*>(
            weight + (size_t)(col_base + n) * IN_F + col_base + c4);
        *reinterpret_cast<float4*>(&wb[n * LDS_STRIDE + c4]) = *src;
    }
    __syncthreads();

    // Wave w computes rows [w*16, w*16+16) x 64 output columns (4 tiles), K=64.
    const int m   = lane & 15;             // row (A) / col (B,D) within tile
    const int khi = (lane >> 4) << 1;      // K sub-offset: 0 (lanes 0-15) or 2

    // Preload all 16 A fragments (K=4 each): lane holds x[row, kA], x[row, kA+1]
    v2f afrag[16];
    #pragma unroll
    for (int kf = 0; kf < 16; ++kf) {
        int kA = kf * 4 + khi;
        afrag[kf] = *reinterpret_cast<const v2f*>(
            &xs[(wave * 16 + m) * LDS_STRIDE + kA]);
    }

    #pragma unroll
    for (int t = 0; t < 4; ++t) {
        v8f acc = {};
        const int n = t * 16 + m;          // B: lane holds column n, rows kB,kB+1
        #pragma unroll
        for (int kf = 0; kf < 16; ++kf) {
            int kB = kf * 4 + khi;
            v2f bfrag = *reinterpret_cast<const v2f*>(
                &wb[n * LDS_STRIDE + kB]);
            // D = A(16x4) * B(4x16) + C, fp32 end to end
            acc = __builtin_amdgcn_wmma_f32_16x16x4_f32(
                false, afrag[kf], false, bfrag, (short)0, acc, false, false);
        }
        // Store D: lane holds column (lane&15); VGPR r -> row r + 8*(lane>>4)
        const int row_off = (lane >> 4) << 3;
        float* outp = out + (size_t)(row_base + wave * 16 + row_off) * IN_F
                          + col_base + t * 16 + m;
        #pragma unroll
        for (int r = 0; r < 8; ++r) {
            outp[(size_t)r * IN_F] = acc[r];
        }
    }
}

extern "C" void kernel_launch(void* const* d_in, const int* in_sizes, int n_in,
                              void* d_out, int out_size, void* d_ws, size_t ws_size,
                              hipStream_t stream) {
    const float* x = (const float*)d_in[0];
    const float* w = (const float*)d_in[1];
    // d_in[2] (mask) is not needed: weight is already block-diagonal and we
    // only read in-block entries where mask == 1.
    float* out = (float*)d_out;

    dim3 grid(N_ROWS / ROWS_PER_WG, 64);   // (64 row tiles, 64 diagonal blocks)
    block_diag_wmma<<<grid, 256, 0, stream>>>(x, w, out);
}